// VoronoiNoise_79654463472357
// MI455X (gfx1250) — compile-verified
//
#include <hip/hip_runtime.h>

typedef __attribute__((ext_vector_type(2))) float v2f;
typedef __attribute__((ext_vector_type(8))) float v8f;

#define VN_W 512
#define VN_H 512
#define VN_B 2
#define VN_NPTS 121
#define VN_NTILE 8   // ceil(121/16), padded points pushed to +1e30 distance^2

// ws layout (uint bit patterns of non-negative floats, so unsigned compare == float compare):
// ws[0]=min b0, ws[1]=max b0, ws[2]=min b1, ws[3]=max b1
__global__ void vor_init(unsigned int* ws) {
    if (threadIdx.x == 0) {
        ws[0] = 0x7F800000u; ws[1] = 0u;
        ws[2] = 0x7F800000u; ws[3] = 0u;
    }
}

// One wave handles 16 consecutive pixels (same row) vs all (padded) 128 points.
// A row (pixel m):  (x, y, 1, x^2+y^2)
// B col (point j):  (-2px, -2py, px^2+py^2 [+pad], 1)
// => (A x B)[m][j] = squared euclidean distance.
__global__ __launch_bounds__(256) void vor_dist(const float* __restrict__ pts,
                                                float* __restrict__ out,
                                                unsigned int* __restrict__ ws) {
    const int lane = threadIdx.x & 31;
    const int g    = blockIdx.x * 8 + (threadIdx.x >> 5);   // pixel-group id (16 px each)
    const int b    = g >> 14;                               // / (512*512/16)
    const int rem  = g & 16383;
    const int y    = rem >> 5;                              // / (512/16)
    const int x0   = (rem & 31) << 4;

    const int  hl = lane & 15;
    const bool hi = (lane >= 16);

    // A matrix: 16x4 f32, 2 VGPRs. lanes 0-15: K=0,1 ; lanes 16-31: K=2,3. M = lane&15.
    const float ax = (float)(x0 + hl);
    const float ay = (float)y;
    v2f a;
    if (!hi) { a.x = ax;   a.y = ay; }
    else     { a.x = 1.0f; a.y = ax * ax + ay * ay; }

    // B matrix tiles: 4x16 f32 each, 2 VGPRs. lanes 0-15: K=0,1 ; lanes 16-31: K=2,3. N = lane&15.
    v2f bt[VN_NTILE];
    const float* pb = pts + (size_t)b * (VN_NPTS * 2);
#pragma unroll
    for (int t = 0; t < VN_NTILE; ++t) {
        int   j  = t * 16 + hl;
        float px = 0.0f, py = 0.0f, pad = 1e30f;
        if (j < VN_NPTS) { px = pb[j * 2]; py = pb[j * 2 + 1]; pad = 0.0f; }
        if (!hi) { bt[t].x = -2.0f * px;           bt[t].y = -2.0f * py; }
        else     { bt[t].x = px * px + py * py + pad; bt[t].y = 1.0f;    }
    }

    // 8x V_WMMA_F32_16X16X4_F32, elementwise min across point tiles.
    v8f zero = {};
    v8f acc = __builtin_amdgcn_wmma_f32_16x16x4_f32(
        false, a, false, bt[0], (short)0, zero, false, false);
#pragma unroll
    for (int t = 1; t < VN_NTILE; ++t) {
        v8f d = __builtin_amdgcn_wmma_f32_16x16x4_f32(
            false, a, false, bt[t], (short)0, zero, false, false);
#pragma unroll
        for (int i = 0; i < 8; ++i) acc[i] = fminf(acc[i], d[i]);
    }

    // Min across the 16 lanes holding point columns (D layout: N = lane&15).
#pragma unroll
    for (int m = 1; m < 16; m <<= 1) {
#pragma unroll
        for (int i = 0; i < 8; ++i)
            acc[i] = fminf(acc[i], __shfl_xor(acc[i], m, 32));
    }
    // Now lanes 0-15 hold minima for rows 0-7 (acc[0..7]); lanes 16-31 rows 8-15.

    // Per-wave min/max -> global atomics (uint-ordered, values >= 0).
    float lm = acc[0], lM = acc[0];
#pragma unroll
    for (int i = 1; i < 8; ++i) { lm = fminf(lm, acc[i]); lM = fmaxf(lM, acc[i]); }
    lm = fminf(lm, __shfl_xor(lm, 16, 32));
    lM = fmaxf(lM, __shfl_xor(lM, 16, 32));
    if (lane == 0) {
        atomicMin(ws + b * 2 + 0, __float_as_uint(sqrtf(fmaxf(lm, 0.0f))));
        atomicMax(ws + b * 2 + 1, __float_as_uint(sqrtf(fmaxf(lM, 0.0f))));
    }

    // Stage per-pixel min distance in out[pixel*3] (channel 0 slot).
    if (lane == 0 || lane == 16) {
        size_t p0 = (size_t)g * 16 + (lane == 16 ? 8 : 0);
#pragma unroll
        for (int i = 0; i < 8; ++i)
            out[(p0 + i) * 3] = sqrtf(fmaxf(acc[i], 0.0f));
    }
}

__global__ __launch_bounds__(256) void vor_norm(float* __restrict__ out,
                                                const unsigned int* __restrict__ ws) {
    size_t p = (size_t)blockIdx.x * 256 + threadIdx.x;      // 0 .. B*H*W-1
    int    b = (int)(p >> 18);                              // / (512*512)
    float lo = __uint_as_float(ws[b * 2 + 0]);
    float hi = __uint_as_float(ws[b * 2 + 1]);
    float m  = out[p * 3];
    float v  = (m - lo) / (hi - lo);
    out[p * 3 + 0] = v;
    out[p * 3 + 1] = v;
    out[p * 3 + 2] = v;
}

extern "C" void kernel_launch(void* const* d_in, const int* in_sizes, int n_in,
                              void* d_out, int out_size, void* d_ws, size_t ws_size,
                              hipStream_t stream) {
    (void)in_sizes; (void)n_in; (void)out_size; (void)ws_size;
    const float*  pts = (const float*)d_in[0];
    float*        out = (float*)d_out;
    unsigned int* ws  = (unsigned int*)d_ws;

    vor_init<<<1, 32, 0, stream>>>(ws);
    // (B*H*W/16) pixel-groups / 8 waves per block = 4096 blocks
    vor_dist<<<4096, 256, 0, stream>>>(pts, out, ws);
    // B*H*W threads
    vor_norm<<<(VN_B * VN_H * VN_W) / 256, 256, 0, stream>>>(out, ws);
}